// MutationGNN_87574383165812
// MI455X (gfx1250) — compile-verified
//
#include <hip/hip_runtime.h>

// GCN (2x GCNConv + FC) for MI455X / gfx1250, wave32.
// Strategy:
//  - degree/rsqrt + edge scatter-add phases are L2-resident (all node arrays
//    ~49MB + 32MB edge list << 192MB L2); bottleneck is f32 atomic throughput,
//    so use 4 threads/edge, float4 gathers, global_atomic_add_f32 scatters.
//  - dense 16x16 and 16x4 GEMMs use V_WMMA_F32_16X16X4_F32 (f32 precision
//    matches the reference; compute is trivial vs the edge phase).
//  - GEMM D-stores take a wave-uniform full-tile fast path (N%16==0 in
//    practice) so no per-store exec masking is emitted.

typedef __attribute__((ext_vector_type(2))) float v2f;
typedef __attribute__((ext_vector_type(8))) float v8f;

#define HID 16
#define NCLS 4

// ---- degree --------------------------------------------------------------
__global__ void k_init_deg(float* __restrict__ deg, int n) {
  int i = blockIdx.x * blockDim.x + threadIdx.x;
  if (i < n) deg[i] = 1.0f;  // self-loop contributes 1 to every node's degree
}

__global__ void k_count_deg(const int* __restrict__ col, float* __restrict__ deg, int e) {
  int i = blockIdx.x * blockDim.x + threadIdx.x;
  if (i < e) atomicAdd(&deg[col[i]], 1.0f);
}

__global__ void k_rsqrt_inplace(float* __restrict__ deg, int n) {
  int i = blockIdx.x * blockDim.x + threadIdx.x;
  if (i < n) {
    float d = deg[i];
    deg[i] = (d > 0.0f) ? rsqrtf(d) : 0.0f;  // deg>=1 always, but keep the guard
  }
}

// ---- layer-1 linear (N x 1 @ 1 x 16) + zero the aggregation buffer -------
__global__ void k_lin1_zero(const float* __restrict__ x, const float* __restrict__ W1,
                            float* __restrict__ h, float* __restrict__ agg, int n) {
  int i = blockIdx.x * blockDim.x + threadIdx.x;
  if (i < n * HID) {
    int node = i >> 4, f = i & 15;
    h[i]   = x[node] * W1[f];
    agg[i] = 0.0f;
  }
}

// ---- edge gather/scale/scatter-add: 4 threads per edge, 4 feats each -----
__global__ void k_edge_scatter(const int* __restrict__ row, const int* __restrict__ col,
                               const float* __restrict__ dinv, const float* __restrict__ h,
                               float* __restrict__ agg, int e) {
  int i = blockIdx.x * blockDim.x + threadIdx.x;
  int tot = e * 4;
  if (i >= tot) return;
  int eidx = i >> 2;
  int q    = i & 3;
  int s = row[eidx];
  int t = col[eidx];
  float w = dinv[s] * dinv[t];
  const float4 v = *(const float4*)(h + (size_t)s * HID + q * 4);
  float* dst = agg + (size_t)t * HID + q * 4;
  atomicAdd(dst + 0, v.x * w);
  atomicAdd(dst + 1, v.y * w);
  atomicAdd(dst + 2, v.z * w);
  atomicAdd(dst + 3, v.w * w);
}

// ---- epilogue: add self-loop message + bias, ReLU; zero h for next layer -
// agg is updated in place (becomes the layer output); h becomes the zeroed
// accumulator for the next aggregation.
__global__ void k_self_bias_relu(float* __restrict__ agg, float* __restrict__ h,
                                 const float* __restrict__ dinv,
                                 const float* __restrict__ bias, int n) {
  int i = blockIdx.x * blockDim.x + threadIdx.x;
  if (i < n * HID) {
    int node = i >> 4, f = i & 15;
    float di = dinv[node];
    float v = agg[i] + h[i] * di * di + bias[f];
    agg[i] = v > 0.0f ? v : 0.0f;
    h[i]   = 0.0f;
  }
}

// ---- WMMA f32 16x16x(4x4) GEMM: out[N,16] = A[N,16] @ W[16,16] ----------
// One wave32 per 16-row tile. A chunk layout (ISA 7.12.2, 32-bit A 16x4):
//   lane<16 -> {A[m][k0],A[m][k0+1]}, lane>=16 -> {A[m][k0+2],A[m][k0+3]}.
// B chunk (4x16, rows striped across lanes): same k-index pattern, col=lane&15.
// C/D: vgpr v, lane<16 -> (M=v, N=lane); lane>=16 -> (M=v+8, N=lane-16).
__global__ void k_gemm16_wmma(const float* __restrict__ A, const float* __restrict__ W,
                              float* __restrict__ out, int n, int ntiles) {
  int wave = (blockIdx.x * blockDim.x + threadIdx.x) >> 5;
  int lane = threadIdx.x & 31;
  if (wave >= ntiles) return;  // wave-uniform: EXEC stays all-ones for WMMA

  const int rbase = wave * 16;
  const int mrow  = lane & 15;
  const int khalf = (lane >> 4) * 2;
  const bool full = (rbase + 16 <= n);  // wave-uniform
  int arow = rbase + mrow;
  if (arow > n - 1) arow = n - 1;  // tail clamp (reads only)

  v8f c = {};
#pragma unroll
  for (int k0 = 0; k0 < 16; k0 += 4) {
    int ka = k0 + khalf;
    v2f a, b;
    a.x = A[(size_t)arow * HID + ka];
    a.y = A[(size_t)arow * HID + ka + 1];
    b.x = W[ka * HID + mrow];
    b.y = W[(ka + 1) * HID + mrow];
    c = __builtin_amdgcn_wmma_f32_16x16x4_f32(false, a, false, b, (short)0, c,
                                              false, false);
  }
  const int mhi = (lane >> 4) * 8;
  if (full) {
    // common path: 8 unguarded stores, no exec masking
#pragma unroll
    for (int v = 0; v < 8; ++v)
      out[(size_t)(rbase + v + mhi) * HID + mrow] = c[v];
  } else {
#pragma unroll
    for (int v = 0; v < 8; ++v) {
      int r = rbase + v + mhi;
      if (r < n) out[(size_t)r * HID + mrow] = c[v];
    }
  }
}

// ---- final FC via WMMA: out[N,4] = A[N,16] @ Wfc[16,4] + bfc -------------
// B is the 16x4 weight zero-padded to 16x16 on the fly (lanes with col>=4
// supply 0); only columns 0..3 of D are stored.
__global__ void k_fc_wmma(const float* __restrict__ A, const float* __restrict__ Wfc,
                          const float* __restrict__ bfc, float* __restrict__ out,
                          int n, int ntiles) {
  int wave = (blockIdx.x * blockDim.x + threadIdx.x) >> 5;
  int lane = threadIdx.x & 31;
  if (wave >= ntiles) return;

  const int rbase = wave * 16;
  const int mrow  = lane & 15;
  const int khalf = (lane >> 4) * 2;
  const bool full = (rbase + 16 <= n);  // wave-uniform
  int arow = rbase + mrow;
  if (arow > n - 1) arow = n - 1;

  v8f c = {};
#pragma unroll
  for (int k0 = 0; k0 < 16; k0 += 4) {
    int ka = k0 + khalf;
    v2f a, b;
    a.x = A[(size_t)arow * HID + ka];
    a.y = A[(size_t)arow * HID + ka + 1];
    b.x = (mrow < NCLS) ? Wfc[ka * NCLS + mrow] : 0.0f;
    b.y = (mrow < NCLS) ? Wfc[(ka + 1) * NCLS + mrow] : 0.0f;
    c = __builtin_amdgcn_wmma_f32_16x16x4_f32(false, a, false, b, (short)0, c,
                                              false, false);
  }
  if (mrow < NCLS) {
    float bias = bfc[mrow];
    const int mhi = (lane >> 4) * 8;
    if (full) {
#pragma unroll
      for (int v = 0; v < 8; ++v)
        out[(size_t)(rbase + v + mhi) * NCLS + mrow] = c[v] + bias;
    } else {
#pragma unroll
      for (int v = 0; v < 8; ++v) {
        int r = rbase + v + mhi;
        if (r < n) out[(size_t)r * NCLS + mrow] = c[v] + bias;
      }
    }
  }
}

extern "C" void kernel_launch(void* const* d_in, const int* in_sizes, int n_in,
                              void* d_out, int out_size, void* d_ws, size_t ws_size,
                              hipStream_t stream) {
  const float* x   = (const float*)d_in[0];
  const int*   ei  = (const int*)d_in[1];   // (2, E): sources then targets
  const float* W1  = (const float*)d_in[2];
  const float* b1  = (const float*)d_in[3];
  const float* W2  = (const float*)d_in[4];
  const float* b2  = (const float*)d_in[5];
  const float* Wfc = (const float*)d_in[6];
  const float* bfc = (const float*)d_in[7];
  float* out = (float*)d_out;

  const int n = in_sizes[0];       // N_NODES (x is N x 1)
  const int e = in_sizes[1] / 2;   // N_EDGES
  const int* row = ei;             // edge_index[0] = sources
  const int* col = ei + e;         // edge_index[1] = targets

  // workspace carve-up: deg/dinv [n] | bufA [n*16] | bufB [n*16] | bufC [n*16]
  float* deg  = (float*)d_ws;
  float* bufA = deg + n;                 // h1, later agg2 -> out2
  float* bufB = bufA + (size_t)n * HID;  // agg1 -> out1
  float* bufC = bufB + (size_t)n * HID;  // h2

  const int T = 256;
  const int gN  = (n + T - 1) / T;
  const int gNF = (n * HID + T - 1) / T;
  const int gE  = (e + T - 1) / T;
  const int gE4 = (e * 4 + T - 1) / T;
  const int ntiles = (n + 15) / 16;
  const int wpb = T / 32;  // waves per block (wave32)
  const int gW  = (ntiles + wpb - 1) / wpb;

  // normalization: deg (with self-loops) -> dinv, in place in `deg`
  k_init_deg     <<<gN, T, 0, stream>>>(deg, n);
  k_count_deg    <<<gE, T, 0, stream>>>(col, deg, e);
  k_rsqrt_inplace<<<gN, T, 0, stream>>>(deg, n);

  // layer 1: h1 = x @ W1 ; agg1 = scatter(norm * h1[row]) ; relu(+b1+self)
  k_lin1_zero    <<<gNF, T, 0, stream>>>(x, W1, bufA, bufB, n);
  k_edge_scatter <<<gE4, T, 0, stream>>>(row, col, deg, bufA, bufB, e);
  k_self_bias_relu<<<gNF, T, 0, stream>>>(bufB, bufA, deg, b1, n);  // bufB=out1, bufA zeroed

  // layer 2: h2 = out1 @ W2 (WMMA) ; aggregate into bufA ; relu(+b2+self)
  k_gemm16_wmma  <<<gW, T, 0, stream>>>(bufB, W2, bufC, n, ntiles);
  k_edge_scatter <<<gE4, T, 0, stream>>>(row, col, deg, bufC, bufA, e);
  k_self_bias_relu<<<gNF, T, 0, stream>>>(bufA, bufC, deg, b2, n);  // bufA=out2

  // classifier: out = out2 @ Wfc + bfc (WMMA, zero-padded B)
  k_fc_wmma      <<<gW, T, 0, stream>>>(bufA, Wfc, bfc, out, n, ntiles);
}